// OperatorModule_73753178407594
// MI455X (gfx1250) — compile-verified
//
#include <hip/hip_runtime.h>
#include <hip/hip_bf16.h>

// Problem constants (match reference)
#define RB   2
#define RH   512
#define RW   512
#define RNA  180
#define RND  512
#define RNT  512

// Padded-image geometry: pad 128 px, row stride 1024 floats, 768 rows.
// Coverage: |coord-255.5| <= 255.5*sqrt2 = 361.4 -> x0 in [-106,617]
// -> padded x0 in [22,745], +1 <= 746 < 1024; rows <= 746 < 768.
#define PPAD  128
#define PW    1024
#define PH    768
#define PIMG  (PH * PW)              // floats per padded image

typedef float v2f __attribute__((ext_vector_type(2)));
typedef float v4f __attribute__((ext_vector_type(4)));
typedef float v8f __attribute__((ext_vector_type(8)));
typedef float f2u __attribute__((ext_vector_type(2), aligned(4)));  // 4B-aligned pair load

// ---------------------------------------------------------------------------
// Prep: zero-padded copy of (B,1,512,512) image into ws (B x 768 x 1024 f32).
// ---------------------------------------------------------------------------
__global__ __launch_bounds__(256)
void pad_image(const float* __restrict__ x, float* __restrict__ ws) {
    int idx = blockIdx.x * 256 + threadIdx.x;           // [0, RB*PIMG)
    int b   = idx / PIMG;
    int r   = idx - b * PIMG;
    int py  = r >> 10;
    int px  = r & (PW - 1);
    int iy  = py - PPAD;
    int ix  = px - PPAD;
    bool valid = ((unsigned)iy < (unsigned)RH) & ((unsigned)ix < (unsigned)RW);
    int yc = iy < 0 ? 0 : (iy > RH - 1 ? RH - 1 : iy);
    int xc = ix < 0 ? 0 : (ix > RW - 1 ? RW - 1 : ix);
    float v = x[(size_t)b * (RH * RW) + yc * RW + xc];
    ws[idx] = valid ? v : 0.0f;
}

// Branch-free bilinear from the padded image: no clamps, no masks.
// 32-bit BYTE offset against a scalar base -> SADDR-form global_load_b64
// (sgpr base + zext(vgpr32)); row y0+1 is the same voffset + 4096B immediate.
__device__ __forceinline__ float bilin_pad(const char* __restrict__ pbase,
                                           float yy, float xx) {
    float fx = floorf(xx);
    float fy = floorf(yy);
    float wx = xx - fx;
    float wy = yy - fy;
    unsigned x0 = (unsigned)(int)fx;
    unsigned y0 = (unsigned)(int)fy;
    unsigned boff = (y0 << 12) | (x0 << 2);      // byte offset: (y0*PW + x0)*4
    f2u p0 = *(const f2u*)(pbase + boff);        // p00, p01
    f2u p1 = *(const f2u*)(pbase + boff + PW * 4);  // p10, p11 (+4096B imm)
    float h0 = fmaf(wx, p0.y - p0.x, p0.x);      // lerp x, row y0
    float h1 = fmaf(wx, p1.y - p1.x, p1.x);      // lerp x, row y0+1
    return fmaf(wy, h1 - h0, h0);                // lerp y
}

// ---------------------------------------------------------------------------
// Main: block = one (batch, angle, group of 8 detector tiles) -> the padded
// image base is uniform (SGPR). Each wave32 owns a 16-detector tile.
//   A (16x4 f32): M = lane&15 (detector); K slots = t offsets
//                 {0,1} on lanes 0-15 (a.x,a.y), {2,3} on lanes 16-31.
//   B = ones(4x16)  =>  D[m,*] accumulates sum over t (full f32 WMMA).
// ---------------------------------------------------------------------------
__global__ __launch_bounds__(256, 8)
void radon_fwd_wmma(const float* __restrict__ ws,
                    const float* __restrict__ angles,
                    float* __restrict__ out) {
    const int lane  = threadIdx.x & 31;
    const int wv    = threadIdx.x >> 5;          // wave in block: 0..7

    const int bat = blockIdx.x / (RNA * 4);      // uniform
    const int rem = blockIdx.x - bat * (RNA * 4);
    const int ang = rem >> 2;                    // uniform
    const int tg  = rem & 3;                     // tile group (uniform)
    const int tile = tg * 8 + wv;                // 0..31

    const float a   = angles[ang];               // scalar load
    const float cv  = cosf(a);
    const float sv  = sinf(a);
    const float nsv = -sv;

    const int   m  = lane & 15;
    const float s  = (float)(tile * 16 + m) - 255.5f;
    // padded-space centers: 255.5 + PPAD = 383.5
    const float bx = fmaf(s, cv, 383.5f);
    const float by = fmaf(s, sv, 383.5f);

    const float koff = (lane < 16) ? 0.0f : 2.0f;
    const float tf0  = koff - 255.5f;            // exact; +4.0 steps exact

    const char* __restrict__ pbase =
        (const char*)(ws + (size_t)bat * PIMG);  // uniform (SGPR) base

    v8f d = {};
    v2f bones;
    bones.x = 1.0f;
    bones.y = 1.0f;

    #pragma unroll 4
    for (int it = 0; it < RNT / 4; ++it) {
        const float tf = tf0 + 4.0f * (float)it;

        v2f av;
        {
            const float xx0 = fmaf(tf, nsv, bx);
            const float yy0 = fmaf(tf, cv,  by);
            av.x = bilin_pad(pbase, yy0, xx0);

            const float t1  = tf + 1.0f;
            const float xx1 = fmaf(t1, nsv, bx);
            const float yy1 = fmaf(t1, cv,  by);
            av.y = bilin_pad(pbase, yy1, xx1);
        }

        d = __builtin_amdgcn_wmma_f32_16x16x4_f32(
                false, av, false, bones, (short)0, d, false, false);
    }

    // D layout: VGPR r -> M=r (lanes 0-15), M=r+8 (lanes 16-31); columns equal.
    float* __restrict__ o = out + ((size_t)bat * RNA + ang) * RND + tile * 16;
    if (lane == 0) {
        v4f lo = {d[0], d[1], d[2], d[3]};
        v4f hi = {d[4], d[5], d[6], d[7]};
        *(v4f*)(o)     = lo;                     // 64B-aligned b128 stores
        *(v4f*)(o + 4) = hi;
    } else if (lane == 16) {
        v4f lo = {d[0], d[1], d[2], d[3]};
        v4f hi = {d[4], d[5], d[6], d[7]};
        *(v4f*)(o + 8)  = lo;
        *(v4f*)(o + 12) = hi;
    }
}

// ---------------------------------------------------------------------------
// Fallback (ws too small): clamped/masked gathers straight from the image.
// ---------------------------------------------------------------------------
__device__ __forceinline__ float tex2d(const float* __restrict__ img, int y, int x) {
    bool valid = ((unsigned)y < (unsigned)RH) & ((unsigned)x < (unsigned)RW);
    int yc = y < 0 ? 0 : (y > RH - 1 ? RH - 1 : y);
    int xc = x < 0 ? 0 : (x > RW - 1 ? RW - 1 : x);
    float p = img[yc * RW + xc];
    return valid ? p : 0.0f;
}

__global__ __launch_bounds__(256, 8)
void radon_fwd_wmma_clamped(const float* __restrict__ x,
                            const float* __restrict__ angles,
                            float* __restrict__ out) {
    const int lane = threadIdx.x & 31;
    const int wv   = threadIdx.x >> 5;
    const int bat  = blockIdx.x / (RNA * 4);
    const int rem  = blockIdx.x - bat * (RNA * 4);
    const int ang  = rem >> 2;
    const int tile = (rem & 3) * 8 + wv;

    const float a   = angles[ang];
    const float cv  = cosf(a);
    const float sv  = sinf(a);
    const float nsv = -sv;

    const int   m  = lane & 15;
    const float s  = (float)(tile * 16 + m) - 255.5f;
    const float bx = fmaf(s, cv, 255.5f);
    const float by = fmaf(s, sv, 255.5f);
    const float koff = (lane < 16) ? 0.0f : 2.0f;
    const float tf0  = koff - 255.5f;

    const float* __restrict__ img = x + (size_t)bat * (RH * RW);

    v8f d = {};
    v2f bones; bones.x = 1.0f; bones.y = 1.0f;

    #pragma unroll 2
    for (int it = 0; it < RNT / 4; ++it) {
        const float tf = tf0 + 4.0f * (float)it;
        v2f av;
        #pragma unroll
        for (int k2 = 0; k2 < 2; ++k2) {
            const float t  = tf + (float)k2;
            const float xx = fmaf(t, nsv, bx);
            const float yy = fmaf(t, cv,  by);
            float fx = floorf(xx), fy = floorf(yy);
            float wx = xx - fx,    wy = yy - fy;
            int x0 = (int)fx, y0 = (int)fy;
            float p00 = tex2d(img, y0,     x0);
            float p01 = tex2d(img, y0,     x0 + 1);
            float p10 = tex2d(img, y0 + 1, x0);
            float p11 = tex2d(img, y0 + 1, x0 + 1);
            float omwx = 1.0f - wx, omwy = 1.0f - wy;
            av[k2] = fmaf(p00, omwy * omwx,
                     fmaf(p01, omwy * wx,
                     fmaf(p10, wy * omwx, p11 * (wy * wx))));
        }
        d = __builtin_amdgcn_wmma_f32_16x16x4_f32(
                false, av, false, bones, (short)0, d, false, false);
    }

    float* __restrict__ o = out + ((size_t)bat * RNA + ang) * RND + tile * 16;
    if (lane == 0) {
        v4f lo = {d[0], d[1], d[2], d[3]};
        v4f hi = {d[4], d[5], d[6], d[7]};
        *(v4f*)(o)     = lo;
        *(v4f*)(o + 4) = hi;
    } else if (lane == 16) {
        v4f lo = {d[0], d[1], d[2], d[3]};
        v4f hi = {d[4], d[5], d[6], d[7]};
        *(v4f*)(o + 8)  = lo;
        *(v4f*)(o + 12) = hi;
    }
}

extern "C" void kernel_launch(void* const* d_in, const int* in_sizes, int n_in,
                              void* d_out, int out_size, void* d_ws, size_t ws_size,
                              hipStream_t stream) {
    (void)in_sizes; (void)n_in; (void)out_size;
    const float* x      = (const float*)d_in[0];   // (B,1,512,512) f32
    const float* angles = (const float*)d_in[1];   // (180,) f32
    float* out          = (float*)d_out;           // (B,1,180,512) f32

    const int nblocks = RB * RNA * 4;              // 1440 blocks x 256 thr
    const size_t need = (size_t)RB * PIMG * sizeof(float);  // 6 MB

    if (ws_size >= need) {
        float* ws = (float*)d_ws;
        const int padblk = (RB * PIMG) / 256;      // 6144 blocks
        pad_image<<<padblk, 256, 0, stream>>>(x, ws);
        radon_fwd_wmma<<<nblocks, 256, 0, stream>>>(ws, angles, out);
    } else {
        radon_fwd_wmma_clamped<<<nblocks, 256, 0, stream>>>(x, angles, out);
    }
}